// LocallyConnectedModel_62182536511570
// MI455X (gfx1250) — compile-verified
//
#include <hip/hip_runtime.h>
#include <hip/hip_bf16.h>

#define W_GRID   128
#define H_GRID   256
#define ITERS    24
#define BATCH    256
#define IN_DIM   784
#define N_CLS    10

typedef __attribute__((ext_vector_type(2))) float v2f;
typedef __attribute__((ext_vector_type(8))) float v8f;

__device__ __forceinline__ float fast_sigmoid(float x) {
    // sigmoid(x) = 1 / (1 + exp(-x)); exp(-x) = exp2(-x * log2(e))
    float e = __builtin_amdgcn_exp2f(x * -1.44269504088896f);
    return __builtin_amdgcn_rcpf(1.0f + e);
}

// ---------------------------------------------------------------------------
// Kernel 1: pre = x(256x784) @ w_pre(784x256) + b_pre, fp32 WMMA 16x16x4.
// One wave32 per 16x16 output tile; 256 tiles -> 32 blocks x 8 waves.
// ---------------------------------------------------------------------------
__global__ __launch_bounds__(256) void pre_gemm_wmma(
        const float* __restrict__ x, const float* __restrict__ w_pre,
        const float* __restrict__ b_pre, float* __restrict__ pre) {
    const int lane = threadIdx.x & 31;
    const int wave = threadIdx.x >> 5;
    const int gw   = blockIdx.x * 8 + wave;       // global wave id 0..255
    const int tm   = (gw >> 4) * 16;              // M tile base
    const int tn   = (gw & 15) * 16;              // N tile base
    const int r    = lane & 15;                   // row/col within tile
    const int kh   = (lane >> 4) * 2;             // K sub-offset (0 or 2)

    v8f c = {};
    for (int k = 0; k < IN_DIM; k += 4) {
        const int kb = k + kh;
        v2f a, bv;
        // A frag: lane holds x[tm+r][kb], x[tm+r][kb+1]
        a.x = x[(tm + r) * IN_DIM + kb];
        a.y = x[(tm + r) * IN_DIM + kb + 1];
        // B frag: lane holds w_pre[kb][tn+r], w_pre[kb+1][tn+r]
        bv.x = w_pre[kb * H_GRID + tn + r];
        bv.y = w_pre[(kb + 1) * H_GRID + tn + r];
        c = __builtin_amdgcn_wmma_f32_16x16x4_f32(false, a, false, bv,
                                                  (short)0, c, false, false);
    }
    const float bias  = b_pre[tn + r];
    const int   mbase = tm + (lane >> 4) * 8;
#pragma unroll
    for (int j = 0; j < 8; ++j)
        pre[(mbase + j) * H_GRID + tn + r] = c[j] + bias;
}

// ---------------------------------------------------------------------------
// Kernel 2: 24 Jacobi iterations of the locally-connected stencil.
// One workgroup (1024 threads = 32 wave32s) per batch element.
// Double-buffered 128x256 fp32 grid in LDS: 2 * 128KB = 256KB of 320KB/WGP.
// Thread t owns cells (w = 4i + t>>8, h = t & 255), i = 0..31:
// neighbor LDS reads are stride-1 across lanes -> bank-conflict free.
// ---------------------------------------------------------------------------
__global__ __launch_bounds__(1024) void stencil_iters(
        const float* __restrict__ pre,
        const float* __restrict__ hw, const float* __restrict__ vw,
        const float* __restrict__ hb, const float* __restrict__ vb,
        float* __restrict__ outrow) {
    extern __shared__ float smem[];               // 2 * 32768 floats
    float* cur = smem;
    float* nxt = smem + (W_GRID * H_GRID);

    const int t  = threadIdx.x;
    const int b  = blockIdx.x;
    const int h  = t & 255;                       // fixed h per thread
    const int tw = t >> 8;                        // 0..3

    // Init: row w==0 gets pre[b][h], everything else 0.
#pragma unroll
    for (int i = 0; i < 32; ++i) {
        const int wv = (i << 2) + tw;
        cur[wv * H_GRID + h] = (wv == 0) ? pre[b * H_GRID + h] : 0.0f;
    }

    // Precompute per-cell bias sums (loop-invariant) into 32 VGPRs.
    float bsum[32];
#pragma unroll
    for (int i = 0; i < 32; ++i) {
        const int wv = (i << 2) + tw;
        float s = 0.0f;
        if (wv > 0)          s += hb[(wv - 1) * H_GRID + h];
        if (wv < W_GRID - 1) s += hb[wv * H_GRID + h];
        if (h > 0)           s += vb[wv * (H_GRID - 1) + (h - 1)];
        if (h < H_GRID - 1)  s += vb[wv * (H_GRID - 1) + h];
        bsum[i] = s;
    }
    __syncthreads();

    for (int it = 0; it < ITERS; ++it) {
#pragma unroll
        for (int i = 0; i < 32; ++i) {
            const int wv = (i << 2) + tw;
            const int c  = wv * H_GRID + h;
            float s = bsum[i];
            if (wv > 0)          s += hw[(wv - 1) * H_GRID + h] * cur[c - H_GRID];
            if (wv < W_GRID - 1) s += hw[wv * H_GRID + h]       * cur[c + H_GRID];
            if (h > 0)           s += vw[wv * (H_GRID - 1) + (h - 1)] * cur[c - 1];
            if (h < H_GRID - 1)  s += vw[wv * (H_GRID - 1) + h]       * cur[c + 1];
            nxt[c] = fast_sigmoid(s);
        }
        __syncthreads();                          // all writes to nxt visible
        float* tmp = cur; cur = nxt; nxt = tmp;   // ping-pong
    }

    // Extract act[:, W_GRID-2, :] for the output projection.
    if (t < H_GRID)
        outrow[b * H_GRID + t] = cur[(W_GRID - 2) * H_GRID + t];
}

// ---------------------------------------------------------------------------
// Kernel 3: out = outrow(256x256) @ w_out(256x10) + b_out, fp32 WMMA.
// 16 M-tiles, single (padded) N-tile of 16 (cols 10..15 are zero).
// One block of 512 threads = 16 wave32s.
// ---------------------------------------------------------------------------
__global__ __launch_bounds__(512) void out_gemm_wmma(
        const float* __restrict__ orow, const float* __restrict__ w_out,
        const float* __restrict__ b_out, float* __restrict__ out) {
    const int lane = threadIdx.x & 31;
    const int wave = threadIdx.x >> 5;            // M tile 0..15
    const int tm   = wave * 16;
    const int r    = lane & 15;
    const int kh   = (lane >> 4) * 2;

    v8f c = {};
    for (int k = 0; k < H_GRID; k += 4) {
        const int kb = k + kh;
        v2f a, bv;
        a.x = orow[(tm + r) * H_GRID + kb];
        a.y = orow[(tm + r) * H_GRID + kb + 1];
        bv.x = (r < N_CLS) ? w_out[kb * N_CLS + r] : 0.0f;
        bv.y = (r < N_CLS) ? w_out[(kb + 1) * N_CLS + r] : 0.0f;
        c = __builtin_amdgcn_wmma_f32_16x16x4_f32(false, a, false, bv,
                                                  (short)0, c, false, false);
    }
    if (r < N_CLS) {
        const float bias  = b_out[r];
        const int   mbase = tm + (lane >> 4) * 8;
#pragma unroll
        for (int j = 0; j < 8; ++j)
            out[(mbase + j) * N_CLS + r] = c[j] + bias;
    }
}

extern "C" void kernel_launch(void* const* d_in, const int* in_sizes, int n_in,
                              void* d_out, int out_size, void* d_ws, size_t ws_size,
                              hipStream_t stream) {
    const float* x     = (const float*)d_in[0];
    const float* w_pre = (const float*)d_in[1];
    const float* b_pre = (const float*)d_in[2];
    const float* hw    = (const float*)d_in[3];
    const float* vw    = (const float*)d_in[4];
    const float* hb    = (const float*)d_in[5];
    const float* vb    = (const float*)d_in[6];
    const float* w_out = (const float*)d_in[7];
    const float* b_out = (const float*)d_in[8];
    float* out  = (float*)d_out;

    float* pre  = (float*)d_ws;                   // 256*256 floats
    float* orow = pre + BATCH * H_GRID;           // 256*256 floats

    pre_gemm_wmma<<<dim3(32), dim3(256), 0, stream>>>(x, w_pre, b_pre, pre);

    const size_t lds_bytes = 2u * W_GRID * H_GRID * sizeof(float); // 256 KB
    stencil_iters<<<dim3(BATCH), dim3(1024), lds_bytes, stream>>>(
        pre, hw, vw, hb, vb, orow);

    out_gemm_wmma<<<dim3(1), dim3(512), 0, stream>>>(orow, w_out, b_out, out);
}